// MaskedCoupling_75419625718213
// MI455X (gfx1250) — compile-verified
//
#include <hip/hip_runtime.h>

typedef __attribute__((ext_vector_type(2))) float v2f;
typedef __attribute__((ext_vector_type(8))) float v8f;

#define D_TOT   1024
#define SPLIT   512
#define HID     128
#define TAILV   3.0f
#define ROWS    32        // batch rows per workgroup (2 row-tiles of 16)
#define PCH     400       // columns per fused chunk = lcm(16,25) -> 16 dims
#define NTILE   25        // 16-col WMMA tiles per chunk
#define NWAVE   8
#define CHW     8         // chunks per wave (4 col-groups * 8 = 32 chunks = 12800 cols)

// LDS layout (in floats).  The h-activation buffers are aliased under the
// param chunk buffers: h2 is hoisted to VGPRs (a-frags) before pb is written.
#define OFF_XT  0
#define OFF_WS  (OFF_XT + ROWS * SPLIT)            // 16384
#define OFF_SH  (OFF_WS + NWAVE * 16)              // 16512
#define SH_FLOATS (NWAVE * 16 * PCH)               // 51200 (> 2*ROWS*HID = 8192)
#define LDS_FLOATS (OFF_SH + SH_FLOATS)            // 67712 floats = 270,848 B

__device__ __forceinline__ float softplus_f(float v) {
  return fmaxf(v, 0.f) + log1pf(__expf(-fabsf(v)));
}

__device__ __forceinline__ v8f wmma_f32(v2f a, v2f b, v8f c) {
  return __builtin_amdgcn_wmma_f32_16x16x4_f32(false, a, false, b, (short)0, c,
                                               false, false);
}

extern "C" __global__ __launch_bounds__(256)
void coupling_fused_kernel(const float* __restrict__ x,
                           const float* __restrict__ W0, const float* __restrict__ b0,
                           const float* __restrict__ W1, const float* __restrict__ b1,
                           const float* __restrict__ W2, const float* __restrict__ b2,
                           const float* __restrict__ W3, const float* __restrict__ b3,
                           float* __restrict__ out_y, float* __restrict__ out_ld)
{
  extern __shared__ float smem[];
  const int tid   = threadIdx.x;
  const int wv    = tid >> 5;
  const int lane  = tid & 31;
  const int mrow  = lane & 15;      // row within 16-row tile / B-frag row / C col
  const int khalf = lane >> 4;      // which K-half of the 16x16x4 fragment
  const int koff  = khalf * 2;
  const int rt    = wv >> 2;        // row-tile (0 or 1) owned by this wave
  const int wg    = wv & 3;         // column group (waves w and w+4 share B loads)
  const int row0  = blockIdx.x * ROWS;
  const int lrow  = rt * 16 + mrow; // local row index (0..31)

  float* xT   = smem + OFF_XT;
  float* wsum = smem + OFF_WS;
  float* hA   = smem + OFF_SH;                 // 32x128
  float* hB   = hA + ROWS * HID;               // 32x128
  float* pb   = smem + OFF_SH + wv * (16 * PCH);  // aliases hA/hB (phase D only)

  // ---- copy x_masked straight to out_y; stage x_transform tile in LDS ----
  for (int e = tid; e < ROWS * SPLIT; e += 256) {
    int r = e >> 9, c = e & (SPLIT - 1);
    const float* xr = x + (size_t)(row0 + r) * D_TOT;
    out_y[(size_t)(row0 + r) * D_TOT + c] = xr[c];
    xT[e] = xr[SPLIT + c];
  }

  // ---- GEMM1: h0 = relu(x_masked @ W0^T + b0); wave does 2 col-tiles ----
#pragma unroll
  for (int tt = 0; tt < 2; ++tt) {
    const int n0 = (wg + tt * 4) * 16;
    v8f acc = {0.f,0.f,0.f,0.f,0.f,0.f,0.f,0.f};
    const float* ap = x  + (size_t)(row0 + lrow) * D_TOT + koff;
    const float* bp = W0 + (size_t)(n0 + mrow) * SPLIT + koff;
#pragma unroll 4
    for (int k = 0; k < SPLIT; k += 4) {
      v2f a = *(const v2f*)(ap + k);
      v2f b = *(const v2f*)(bp + k);
      acc = wmma_f32(a, b, acc);
    }
    float bias = b0[n0 + mrow];
#pragma unroll
    for (int r = 0; r < 8; ++r)
      hA[(rt * 16 + r + khalf * 8) * HID + n0 + mrow] = fmaxf(acc[r] + bias, 0.f);
  }
  __syncthreads();

  // ---- GEMM2: h1 = relu(h0 @ W1^T + b1) ----
#pragma unroll
  for (int tt = 0; tt < 2; ++tt) {
    const int n0 = (wg + tt * 4) * 16;
    v8f acc = {0.f,0.f,0.f,0.f,0.f,0.f,0.f,0.f};
    const float* ap = hA + lrow * HID + koff;
    const float* bp = W1 + (size_t)(n0 + mrow) * HID + koff;
#pragma unroll
    for (int k = 0; k < HID; k += 4) {
      v2f a = *(const v2f*)(ap + k);
      v2f b = *(const v2f*)(bp + k);
      acc = wmma_f32(a, b, acc);
    }
    float bias = b1[n0 + mrow];
#pragma unroll
    for (int r = 0; r < 8; ++r)
      hB[(rt * 16 + r + khalf * 8) * HID + n0 + mrow] = fmaxf(acc[r] + bias, 0.f);
  }
  __syncthreads();

  // ---- GEMM3: h2 = relu(h1 @ W2^T + b2), result back into hA ----
#pragma unroll
  for (int tt = 0; tt < 2; ++tt) {
    const int n0 = (wg + tt * 4) * 16;
    v8f acc = {0.f,0.f,0.f,0.f,0.f,0.f,0.f,0.f};
    const float* ap = hB + lrow * HID + koff;
    const float* bp = W2 + (size_t)(n0 + mrow) * HID + koff;
#pragma unroll
    for (int k = 0; k < HID; k += 4) {
      v2f a = *(const v2f*)(ap + k);
      v2f b = *(const v2f*)(bp + k);
      acc = wmma_f32(a, b, acc);
    }
    float bias = b2[n0 + mrow];
#pragma unroll
    for (int r = 0; r < 8; ++r)
      hA[(rt * 16 + r + khalf * 8) * HID + n0 + mrow] = fmaxf(acc[r] + bias, 0.f);
  }
  __syncthreads();

  // ---- hoist this wave's h2 A-fragments (16x128) into 64 VGPRs; after the
  //      barrier the hA/hB space is re-used as the param chunk buffers ----
  v2f afrag[32];
#pragma unroll
  for (int kk = 0; kk < 32; ++kk)
    afrag[kk] = *(const v2f*)(hA + lrow * HID + kk * 4 + koff);
  __syncthreads();

  float ldpart = 0.f;

  // ---- GEMM4 + RQS, fully fused. Wave pairs (w, w+4) traverse identical W3
  //      columns for different row-tiles -> WGP$ halves the L2 traffic. The
  //      per-chunk barrier keeps pair members in lockstep for cache reuse. ----
  for (int cc = 0; cc < CHW; ++cc) {
    const int chunk = wg + 4 * cc;          // 0..31, 400 cols / 16 dims each
    const int col0  = chunk * PCH;

    // 25 WMMA column-tiles -> per-wave LDS param buffer
    for (int t = 0; t < NTILE; ++t) {
      const int n0c = col0 + t * 16;
      v8f acc = {0.f,0.f,0.f,0.f,0.f,0.f,0.f,0.f};
      const float* bp = W3 + (size_t)(n0c + mrow) * HID + koff;
#pragma unroll
      for (int kk = 0; kk < 32; ++kk) {
        v2f b = *(const v2f*)(bp + kk * 4);
        acc = wmma_f32(afrag[kk], b, acc);
      }
      float bias = b3[n0c + mrow];
#pragma unroll
      for (int r = 0; r < 8; ++r)
        pb[(r + khalf * 8) * PCH + t * 16 + mrow] = acc[r] + bias;
    }

    // RQS spline for the 16 dims of this chunk.
    // Lane handles row=lrow, dims dl = khalf + 2*i (i = 0..7).
#pragma unroll 1
    for (int i = 0; i < 8; ++i) {
      const int dl   = khalf + 2 * i;
      const int gdim = chunk * 16 + dl;
      const float* p = pb + mrow * PCH + dl * 25;

      float pw[8], ph[8], dv[9];
      float mw = -1e30f, mh = -1e30f;
#pragma unroll
      for (int j = 0; j < 8; ++j) {
        pw[j] = p[j];      mw = fmaxf(mw, pw[j]);
        ph[j] = p[8 + j];  mh = fmaxf(mh, ph[j]);
      }
      float sw = 0.f, sh = 0.f;
#pragma unroll
      for (int j = 0; j < 8; ++j) {
        pw[j] = __expf(pw[j] - mw); sw += pw[j];
        ph[j] = __expf(ph[j] - mh); sh += ph[j];
      }
      const float scw = (2.0f * TAILV) / sw;
      const float sch = (2.0f * TAILV) / sh;
#pragma unroll
      for (int j = 0; j < 8; ++j) { pw[j] *= scw; ph[j] *= sch; }
#pragma unroll
      for (int j = 0; j < 9; ++j) dv[j] = softplus_f(p[16 + j]);

      const float xv     = xT[lrow * SPLIT + gdim];
      const bool  inside = (xv > -TAILV) && (xv < TAILV);
      const float xc     = fminf(fmaxf(xv, -TAILV), TAILV);

      // bin select: last j in [0,7] with xc >= cumwidth[j]
      float cw = -TAILV, chh = -TAILV;
      float x_k = -TAILV, y_k = -TAILV;
      float w_k = pw[0], h_k = ph[0], d_k = dv[0], d_k1 = dv[1];
#pragma unroll
      for (int j = 0; j < 8; ++j) {
        bool sel = (xc >= cw);
        if (sel) { x_k = cw; y_k = chh; w_k = pw[j]; h_k = ph[j];
                   d_k = dv[j]; d_k1 = dv[j + 1]; }
        cw += pw[j]; chh += ph[j];
      }

      const float s    = h_k / w_k;
      const float th   = (xc - x_k) / w_k;
      const float omt  = 1.f - th;
      const float den  = s + (d_k1 + d_k - 2.f * s) * th * omt;
      const float y_in = y_k + h_k * (s * th * th + d_k * th * omt) / den;
      const float num  = s * s * (d_k1 * th * th + 2.f * s * th * omt + d_k * omt * omt);
      const float ld_in = __logf(num) - 2.f * __logf(den);

      out_y[(size_t)(row0 + lrow) * D_TOT + SPLIT + gdim] = inside ? y_in : xv;
      ldpart += inside ? ld_in : 0.f;
    }
    __syncthreads();   // bound the skew between wave pairs sharing W3 via WGP$
  }

  // ---- deterministic logdet reduction ----
  ldpart += __shfl_xor(ldpart, 16, 32);   // combine the two K-halves (same row)
  if (lane < 16) wsum[wv * 16 + lane] = ldpart;
  __syncthreads();

  if (tid < ROWS) {
    const int r2 = tid >> 4;              // row-tile of this row
    float s = 0.f;
#pragma unroll
    for (int g = 0; g < 4; ++g) s += wsum[(r2 * 4 + g) * 16 + (tid & 15)];
    out_ld[row0 + tid] = s;
  }
}

extern "C" void kernel_launch(void* const* d_in, const int* in_sizes, int n_in,
                              void* d_out, int out_size, void* d_ws, size_t ws_size,
                              hipStream_t stream) {
  (void)in_sizes; (void)n_in; (void)d_ws; (void)ws_size; (void)out_size;
  const float* x  = (const float*)d_in[0];
  const float* W0 = (const float*)d_in[1];
  const float* b0 = (const float*)d_in[2];
  const float* W1 = (const float*)d_in[3];
  const float* b1 = (const float*)d_in[4];
  const float* W2 = (const float*)d_in[5];
  const float* b2 = (const float*)d_in[6];
  const float* W3 = (const float*)d_in[7];
  const float* b3 = (const float*)d_in[8];

  float* out_y  = (float*)d_out;
  float* out_ld = out_y + (size_t)8192 * 1024;

  const size_t lds_bytes = (size_t)LDS_FLOATS * sizeof(float); // ~271 KB < 320 KB
  dim3 grid(8192 / ROWS), block(256);
  coupling_fused_kernel<<<grid, block, lds_bytes, stream>>>(
      x, W0, b0, W1, b1, W2, b2, W3, b3, out_y, out_ld);
}